// PatchLoss_16690242912806
// MI455X (gfx1250) — compile-verified
//
#include <hip/hip_runtime.h>

typedef __attribute__((ext_vector_type(2))) float v2f;
typedef __attribute__((ext_vector_type(4))) float v4f;
typedef __attribute__((ext_vector_type(8))) float v8f;

#define IMG_W 8192L
#define GROUPS_PER_WAVE 8
#define WAVES_PER_BLOCK 8

__global__ void PatchLoss_init_kernel(float* out) {
    out[0] = 0.0f;  // max starts at 0 => implements jnp.maximum(max, 0)
}

// One wave (32 lanes) owns a group of 16 horizontally-adjacent 8x8 patches.
// Lane pair (m, m+16) feeds patch m: lane m supplies pixel rows 0..3, lane
// m+16 supplies rows 4..7 (ISA A-layout: lanes 0-15 -> K=0,1 of row M=lane;
// lanes 16-31 -> K=2,3 of row M=lane-16). With B == all-ones,
// D[m][n] = sum_k A[m][k] + C[m][n], so 16 chained WMMAs accumulate the full
// 64-element |o-t| sum of all 16 patches into the f32 accumulator.
__global__ __launch_bounds__(256) void PatchLoss_16690242912806_kernel(
    const float* __restrict__ o, const float* __restrict__ t,
    float* __restrict__ out) {
    const int tid = threadIdx.x;
    const int lane = tid & 31;
    const int waveInBlock = tid >> 5;
    const int waveId = blockIdx.x * WAVES_PER_BLOCK + waveInBlock;
    const int lp = lane & 15;   // patch-in-group
    const int half = lane >> 4; // 0: pixel rows 0-3, 1: pixel rows 4-7

    v2f ones;
    ones.x = 1.0f;
    ones.y = 1.0f;

    float wavemax = 0.0f;

    for (int gi = 0; gi < GROUPS_PER_WAVE; ++gi) {
        const int g = waveId * GROUPS_PER_WAVE + gi; // 0..65535
        const int prow = g >> 6;                     // patch row 0..1023
        const int gcol = g & 63;                     // group-of-16 column
        const long col0 = ((long)(gcol * 16 + lp)) * 8;      // pixel col
        const long row0 = (long)prow * 8 + (long)half * 4;   // pixel row

        v8f acc = {}; // C = 0
#pragma unroll
        for (int r = 0; r < 4; ++r) {
            const float* op = o + (row0 + r) * IMG_W + col0;
            const float* tp = t + (row0 + r) * IMG_W + col0;
            v4f o0 = __builtin_nontemporal_load((const v4f*)op);
            v4f o1 = __builtin_nontemporal_load((const v4f*)(op + 4));
            v4f t0 = __builtin_nontemporal_load((const v4f*)tp);
            v4f t1 = __builtin_nontemporal_load((const v4f*)(tp + 4));

            v2f a;
            a.x = __builtin_fabsf(o0.x - t0.x);
            a.y = __builtin_fabsf(o0.y - t0.y);
            acc = __builtin_amdgcn_wmma_f32_16x16x4_f32(false, a, false, ones,
                                                        (short)0, acc, false, false);
            a.x = __builtin_fabsf(o0.z - t0.z);
            a.y = __builtin_fabsf(o0.w - t0.w);
            acc = __builtin_amdgcn_wmma_f32_16x16x4_f32(false, a, false, ones,
                                                        (short)0, acc, false, false);
            a.x = __builtin_fabsf(o1.x - t1.x);
            a.y = __builtin_fabsf(o1.y - t1.y);
            acc = __builtin_amdgcn_wmma_f32_16x16x4_f32(false, a, false, ones,
                                                        (short)0, acc, false, false);
            a.x = __builtin_fabsf(o1.z - t1.z);
            a.y = __builtin_fabsf(o1.w - t1.w);
            acc = __builtin_amdgcn_wmma_f32_16x16x4_f32(false, a, false, ones,
                                                        (short)0, acc, false, false);
        }
        // Each lane holds 8 patch sums (replicated over 16 lanes); max over them.
        float gm = acc[0];
#pragma unroll
        for (int v = 1; v < 8; ++v) gm = __builtin_fmaxf(gm, acc[v]);
        wavemax = __builtin_fmaxf(wavemax, gm);
    }

    // wave32 max reduction
#pragma unroll
    for (int off = 16; off > 0; off >>= 1)
        wavemax = __builtin_fmaxf(wavemax, __shfl_xor(wavemax, off, 32));

    __shared__ float smax[WAVES_PER_BLOCK];
    if (lane == 0) smax[waveInBlock] = wavemax;
    __syncthreads();
    if (tid == 0) {
        float bm = smax[0];
#pragma unroll
        for (int i = 1; i < WAVES_PER_BLOCK; ++i) bm = __builtin_fmaxf(bm, smax[i]);
        bm *= (1.0f / 64.0f); // per-patch mean = sum/64; scale commutes with max
        // MAE >= 0, so float bits compare monotonically as unsigned.
        atomicMax((unsigned int*)out, __float_as_uint(bm));
    }
}

extern "C" void kernel_launch(void* const* d_in, const int* in_sizes, int n_in,
                              void* d_out, int out_size, void* d_ws, size_t ws_size,
                              hipStream_t stream) {
    (void)in_sizes; (void)n_in; (void)d_ws; (void)ws_size; (void)out_size;
    const float* o = (const float*)d_in[0];
    const float* t = (const float*)d_in[1];
    float* out = (float*)d_out;

    PatchLoss_init_kernel<<<1, 1, 0, stream>>>(out);

    // 1024x1024 patches -> 65536 groups of 16 -> 8192 waves -> 1024 blocks.
    const int totalGroups = (1024 * 1024) / 16;
    const int totalWaves = totalGroups / GROUPS_PER_WAVE;      // 8192
    const int blocks = totalWaves / WAVES_PER_BLOCK;           // 1024
    PatchLoss_16690242912806_kernel<<<blocks, 32 * WAVES_PER_BLOCK, 0, stream>>>(o, t, out);
}